// PointNetSetAbstraction_42185168781351
// MI455X (gfx1250) — compile-verified
//
#include <hip/hip_runtime.h>
#include <hip/hip_bf16.h>

#define B_ 16
#define N_ 4096
#define S_ 1024
#define K_ 32
#define M_ (B_ * S_ * K_)   // 524288 rows
#define EPS_ 1e-5f

typedef __attribute__((ext_vector_type(2))) float v2f;
typedef __attribute__((ext_vector_type(8))) float v8f;

// ---------------------------------------------------------------------------
// Farthest point sampling: one block (1024 threads) per batch.
// Each thread owns 4 points in registers. Writes new_xyz (B,3,S) directly
// into the first output region.
// ---------------------------------------------------------------------------
__global__ __launch_bounds__(1024) void fps_kernel(const float* __restrict__ xyz,
                                                   float* __restrict__ out_newxyz) {
  const int b = blockIdx.x;
  const int tid = threadIdx.x;
  const float* X = xyz + (size_t)b * 3 * N_;
  float px[4], py[4], pz[4], dist[4];
#pragma unroll
  for (int p = 0; p < 4; ++p) {
    int n = tid + p * 1024;
    px[p] = X[n];
    py[p] = X[N_ + n];
    pz[p] = X[2 * N_ + n];
    dist[p] = 1e10f;
  }
  __shared__ float s_c[3];
  __shared__ float s_v[32];
  __shared__ int s_i[32];
  __shared__ int s_far;
  int far = 0;
  for (int s = 0; s < S_; ++s) {
    int own_t = far & 1023;
    int own_p = far >> 10;
    if (tid == own_t) {
      float cx = px[own_p], cy = py[own_p], cz = pz[own_p];
      s_c[0] = cx; s_c[1] = cy; s_c[2] = cz;
      out_newxyz[(b * 3 + 0) * S_ + s] = cx;
      out_newxyz[(b * 3 + 1) * S_ + s] = cy;
      out_newxyz[(b * 3 + 2) * S_ + s] = cz;
    }
    __syncthreads();
    float cx = s_c[0], cy = s_c[1], cz = s_c[2];
    float bv = -1.0f;
    int bi = 0;
#pragma unroll
    for (int p = 0; p < 4; ++p) {
      float dx = px[p] - cx, dy = py[p] - cy, dz = pz[p] - cz;
      float dd = dx * dx + dy * dy + dz * dz;
      float nd = fminf(dist[p], dd);
      dist[p] = nd;
      if (nd > bv) { bv = nd; bi = tid + p * 1024; }
    }
#pragma unroll
    for (int off = 16; off > 0; off >>= 1) {
      float ov = __shfl_xor(bv, off, 32);
      int oi = __shfl_xor(bi, off, 32);
      if (ov > bv || (ov == bv && oi < bi)) { bv = ov; bi = oi; }
    }
    if ((tid & 31) == 0) { s_v[tid >> 5] = bv; s_i[tid >> 5] = bi; }
    __syncthreads();
    if (tid < 32) {
      bv = s_v[tid];
      bi = s_i[tid];
#pragma unroll
      for (int off = 16; off > 0; off >>= 1) {
        float ov = __shfl_xor(bv, off, 32);
        int oi = __shfl_xor(bi, off, 32);
        if (ov > bv || (ov == bv && oi < bi)) { bv = ov; bi = oi; }
      }
      if (tid == 0) s_far = bi;
    }
    __syncthreads();
    far = s_far;
  }
}

// ---------------------------------------------------------------------------
// Transpose points (B,64,N) -> (B,N,64) so grouping reads are coalesced.
// ---------------------------------------------------------------------------
__global__ __launch_bounds__(256) void transpose_pts(const float* __restrict__ pts,
                                                     float* __restrict__ ptst) {
  long e = (long)blockIdx.x * 256 + threadIdx.x;   // < B*64*N
  int b = (int)(e >> 18);                          // 64*4096 = 2^18
  int r = (int)(e & 262143);
  int c = r >> 12;
  int n = r & 4095;
  ptst[((long)b * N_ + n) * 64 + c] = pts[e];
}

// ---------------------------------------------------------------------------
// Ball query: one wave per (b,s). Takes the first K in-radius indices in
// index order (== top_k of -gid in the reference), pads with the first hit.
// ---------------------------------------------------------------------------
__global__ __launch_bounds__(256) void ballquery_kernel(const float* __restrict__ xyz,
                                                        const float* __restrict__ newxyz,
                                                        int* __restrict__ ball_idx) {
  const float R2 = 0.2f * 0.2f;
  int wid = (int)(((long)blockIdx.x * 256 + threadIdx.x) >> 5);  // 0..B*S-1
  int lane = threadIdx.x & 31;
  int b = wid >> 10;
  int s = wid & 1023;
  const float* X = xyz + (size_t)b * 3 * N_;
  float cx = newxyz[(b * 3 + 0) * S_ + s];
  float cy = newxyz[(b * 3 + 1) * S_ + s];
  float cz = newxyz[(b * 3 + 2) * S_ + s];
  float sumc = cx * cx + cy * cy + cz * cz;
  int* out = ball_idx + (size_t)wid * K_;
  int cnt = 0;
  int first = 0;
  bool have_first = false;
  for (int n0 = 0; n0 < N_ && cnt < K_; n0 += 32) {
    int n = n0 + lane;
    float x = X[n], y = X[N_ + n], z = X[2 * N_ + n];
    float dot = cx * x + cy * y + cz * z;
    float sq = sumc - 2.0f * dot + (x * x + y * y + z * z);
    bool pred = !(sq > R2);
    unsigned mask = __builtin_amdgcn_ballot_w32(pred);
    if (mask) {
      if (!have_first) { first = n0 + __builtin_ctz(mask); have_first = true; }
      int pos = __popc(mask & ((1u << lane) - 1u));
      if (pred && (cnt + pos) < K_) out[cnt + pos] = n;
      cnt += __popc(mask);
    }
  }
  if (cnt < K_) {
    for (int slot = cnt + lane; slot < K_; slot += 32) out[slot] = first;
  }
}

// ---------------------------------------------------------------------------
// Grouping: one wave per output row m=(b,s,k). Builds feat0[m][0..67]:
// channels 0..2 = grouped_xyz - center, 3..66 = grouped points, 67 = 0 pad.
// ---------------------------------------------------------------------------
__global__ __launch_bounds__(256) void group_kernel(const float* __restrict__ xyz,
                                                    const float* __restrict__ newxyz,
                                                    const float* __restrict__ ptst,
                                                    const int* __restrict__ ball_idx,
                                                    float* __restrict__ feat0) {
  int wid = (int)(((long)blockIdx.x * 256 + threadIdx.x) >> 5);  // row m
  int lane = threadIdx.x & 31;
  int b = wid >> 15;          // S*K = 32768
  int rem = wid & 32767;
  int s = rem >> 5;
  int idxn = ball_idx[wid];
  const float* prow = ptst + ((long)b * N_ + idxn) * 64;
  float* frow = feat0 + (long)wid * 68;
  float v0;
  if (lane < 3)
    v0 = xyz[((long)b * 3 + lane) * N_ + idxn] - newxyz[(b * 3 + lane) * S_ + s];
  else
    v0 = prow[lane - 3];
  frow[lane] = v0;
  frow[lane + 32] = prow[lane + 29];
  if (lane < 4) frow[lane + 64] = (lane < 3) ? prow[lane + 61] : 0.0f;
}

// ---------------------------------------------------------------------------
// WMMA f32 GEMM: Out[M,COUT] = act(A[M,CIN]) * W^T + bias,
// plus deterministic per-block partial sum/sumsq for BN statistics.
// Block = 8 waves, each wave a 16-row tile; wave covers all COUT via
// v_wmma_f32_16x16x4_f32 with K stepped by 4. Weights live in LDS.
// If BN_IN: input is previous pre-activation, normalize+ReLU fused on load.
// ---------------------------------------------------------------------------
template <int CIN, int KDIM, int CIN_PAD, int COUT, bool BN_IN>
__global__ __launch_bounds__(256) void gemm_bn_kernel(const float* __restrict__ Ain,
                                                      const float* __restrict__ W,
                                                      const float* __restrict__ bias,
                                                      const float* __restrict__ scale_in,
                                                      const float* __restrict__ shift_in,
                                                      float* __restrict__ Out,
                                                      float* __restrict__ partial) {
  __shared__ float lw[COUT * CIN_PAD];
  __shared__ float lbias[COUT];
  __shared__ float lscale[64];
  __shared__ float lshift[64];
  __shared__ float rsum[8][COUT];
  __shared__ float rsq[8][COUT];

  const int tid = threadIdx.x;
  for (int i = tid; i < COUT * CIN_PAD; i += 256) {
    int o = i / CIN_PAD;
    int c = i - o * CIN_PAD;
    lw[i] = (c < CIN) ? W[o * CIN + c] : 0.0f;
  }
  for (int i = tid; i < COUT; i += 256) lbias[i] = bias[i];
  if constexpr (BN_IN) {
    for (int i = tid; i < CIN; i += 256) {
      lscale[i] = scale_in[i];
      lshift[i] = shift_in[i];
    }
  }
  __syncthreads();

  const int wave = tid >> 5;
  const int lane = tid & 31;
  const int lo = lane & 15;
  const int hi = lane >> 4;
  const long m_base = (long)blockIdx.x * 128 + wave * 16;
  const long mrow = m_base + lo;
  const int NT = COUT / 16;

  v8f acc[NT];
  const v8f zero = {0.f, 0.f, 0.f, 0.f, 0.f, 0.f, 0.f, 0.f};
#pragma unroll
  for (int t = 0; t < NT; ++t) acc[t] = zero;

  const float* arow = Ain + mrow * (long)CIN_PAD;
  for (int k0 = 0; k0 < KDIM; k0 += 4) {
    int c = k0 + hi * 2;
    v2f a = *(const v2f*)(arow + c);
    if constexpr (BN_IN) {
      a.x = fmaxf(fmaf(a.x, lscale[c], lshift[c]), 0.0f);
      a.y = fmaxf(fmaf(a.y, lscale[c + 1], lshift[c + 1]), 0.0f);
    }
#pragma unroll
    for (int t = 0; t < NT; ++t) {
      int o = t * 16 + lo;
      v2f bb = *(const v2f*)&lw[o * CIN_PAD + c];
      acc[t] = __builtin_amdgcn_wmma_f32_16x16x4_f32(false, a, false, bb, (short)0,
                                                     acc[t], false, false);
    }
  }

  // epilogue: add bias, store, accumulate per-channel partial statistics
#pragma unroll
  for (int t = 0; t < NT; ++t) {
    int o = t * 16 + lo;
    float bo = lbias[o];
    float ps = 0.0f, pq = 0.0f;
#pragma unroll
    for (int j = 0; j < 8; ++j) {
      float v = acc[t][j] + bo;
      long mm = m_base + j + 8 * hi;
      Out[mm * COUT + o] = v;
      ps += v;
      pq += v * v;
    }
    ps += __shfl_xor(ps, 16, 32);
    pq += __shfl_xor(pq, 16, 32);
    if (hi == 0) { rsum[wave][o] = ps; rsq[wave][o] = pq; }
  }
  __syncthreads();
  if (tid < COUT) {
    float s = 0.0f, q = 0.0f;
#pragma unroll
    for (int w = 0; w < 8; ++w) { s += rsum[w][tid]; q += rsq[w][tid]; }
    partial[(long)blockIdx.x * (2 * COUT) + tid] = s;
    partial[(long)blockIdx.x * (2 * COUT) + COUT + tid] = q;
  }
}

// ---------------------------------------------------------------------------
// Reduce per-block partials into BN scale/shift (deterministic tree).
// scale = g * rsqrt(var+eps); shift = be - scale * mean
// ---------------------------------------------------------------------------
__global__ __launch_bounds__(256) void bn_finalize_kernel(const float* __restrict__ partial,
                                                          int nblocks, int cout,
                                                          const float* __restrict__ g,
                                                          const float* __restrict__ be,
                                                          float* __restrict__ scale_out,
                                                          float* __restrict__ shift_out) {
  int c = blockIdx.x;
  int tid = threadIdx.x;
  __shared__ float ss[256];
  __shared__ float sq[256];
  float s = 0.0f, q = 0.0f;
  for (int i = tid; i < nblocks; i += 256) {
    s += partial[(long)i * (2 * cout) + c];
    q += partial[(long)i * (2 * cout) + cout + c];
  }
  ss[tid] = s;
  sq[tid] = q;
  __syncthreads();
  for (int off = 128; off > 0; off >>= 1) {
    if (tid < off) { ss[tid] += ss[tid + off]; sq[tid] += sq[tid + off]; }
    __syncthreads();
  }
  if (tid == 0) {
    const float inv_m = 1.0f / (float)M_;
    float mean = ss[0] * inv_m;
    float var = sq[0] * inv_m - mean * mean;
    float inv = rsqrtf(var + EPS_);
    float sc = g[c] * inv;
    scale_out[c] = sc;
    shift_out[c] = be[c] - sc * mean;
  }
}

// ---------------------------------------------------------------------------
// Final BN + ReLU + max over K, writing (B,128,S).
// ---------------------------------------------------------------------------
__global__ __launch_bounds__(128) void maxpool_kernel(const float* __restrict__ pre3,
                                                      const float* __restrict__ scale,
                                                      const float* __restrict__ shift,
                                                      float* __restrict__ out2) {
  int bs = blockIdx.x;   // 0..B*S-1
  int o = threadIdx.x;   // 0..127
  int b = bs >> 10;
  int s = bs & 1023;
  const float* base = pre3 + (long)bs * K_ * 128 + o;
  float sc = scale[o], sh = shift[o];
  float m = 0.0f;  // ReLU outputs are >= 0
#pragma unroll 4
  for (int k = 0; k < K_; ++k)
    m = fmaxf(m, fmaxf(fmaf(base[k * 128], sc, sh), 0.0f));
  out2[((long)b * 128 + o) * S_ + s] = m;
}

// ---------------------------------------------------------------------------
extern "C" void kernel_launch(void* const* d_in, const int* in_sizes, int n_in,
                              void* d_out, int out_size, void* d_ws, size_t ws_size,
                              hipStream_t stream) {
  (void)in_sizes; (void)n_in; (void)out_size; (void)ws_size;
  const float* xyz = (const float*)d_in[0];
  const float* pts = (const float*)d_in[1];
  const float* w0 = (const float*)d_in[2];
  const float* b0 = (const float*)d_in[3];
  const float* g0 = (const float*)d_in[4];
  const float* be0 = (const float*)d_in[5];
  const float* w1 = (const float*)d_in[6];
  const float* b1 = (const float*)d_in[7];
  const float* g1 = (const float*)d_in[8];
  const float* be1 = (const float*)d_in[9];
  const float* w2 = (const float*)d_in[10];
  const float* b2 = (const float*)d_in[11];
  const float* g2 = (const float*)d_in[12];
  const float* be2 = (const float*)d_in[13];

  float* out = (float*)d_out;
  float* out_xyz = out;                  // (B,3,S)
  float* out_pts = out + B_ * 3 * S_;    // (B,128,S)

  char* ws = (char*)d_ws;
  int* d_ball = (int*)ws;                                   // 2 MB
  float* d_part = (float*)(ws + (2ull << 20));              // 4 MB
  float* d_stats = (float*)(ws + (6ull << 20));             // 4 KB
  float* d_ptst = (float*)(ws + (6ull << 20) + 4096);       // 16 MB
  float* d_feat0 = (float*)(ws + (22ull << 20) + 4096);     // M*68 f32
  float* d_pre1 = d_feat0 + (size_t)M_ * 68;                // M*64 f32
  float* d_pre2 = d_feat0;                                  // reuse feat0 region
  float* d_pre3 = d_pre1;                                   // M*128 f32 (extends past pre1)

  float* sc0 = d_stats + 0,   *sh0 = d_stats + 128;
  float* sc1 = d_stats + 256, *sh1 = d_stats + 384;
  float* sc2 = d_stats + 512, *sh2 = d_stats + 640;

  fps_kernel<<<B_, 1024, 0, stream>>>(xyz, out_xyz);
  transpose_pts<<<(B_ * 64 * N_) / 256, 256, 0, stream>>>(pts, d_ptst);
  ballquery_kernel<<<(B_ * S_) / 8, 256, 0, stream>>>(xyz, out_xyz, d_ball);
  group_kernel<<<M_ / 8, 256, 0, stream>>>(xyz, out_xyz, d_ptst, d_ball, d_feat0);

  gemm_bn_kernel<67, 68, 68, 64, false>
      <<<M_ / 128, 256, 0, stream>>>(d_feat0, w0, b0, nullptr, nullptr, d_pre1, d_part);
  bn_finalize_kernel<<<64, 256, 0, stream>>>(d_part, M_ / 128, 64, g0, be0, sc0, sh0);

  gemm_bn_kernel<64, 64, 64, 64, true>
      <<<M_ / 128, 256, 0, stream>>>(d_pre1, w1, b1, sc0, sh0, d_pre2, d_part);
  bn_finalize_kernel<<<64, 256, 0, stream>>>(d_part, M_ / 128, 64, g1, be1, sc1, sh1);

  gemm_bn_kernel<64, 64, 64, 128, true>
      <<<M_ / 128, 256, 0, stream>>>(d_pre2, w2, b2, sc1, sh1, d_pre3, d_part);
  bn_finalize_kernel<<<128, 256, 0, stream>>>(d_part, M_ / 128, 128, g2, be2, sc2, sh2);

  maxpool_kernel<<<B_ * S_, 128, 0, stream>>>(d_pre3, sc2, sh2, out_pts);
}